// SRTM_90503550861328
// MI455X (gfx1250) — compile-verified
//
#include <hip/hip_runtime.h>
#include <hip/hip_bf16.h>

typedef float v2f __attribute__((ext_vector_type(2)));
typedef float v8f __attribute__((ext_vector_type(8)));

#define T_FRAMES   61
#define DT         (1.0f / 60.0f)
#define WG_THREADS 256
#define VOX_PER_WG 128          // 8 wave32 * 16 voxel columns each
#define FS         66           // F LDS row stride (floats), keeps float2 loads 8B-aligned
#define SS         130          // S LDS row stride (floats), bank-conflict-free stores/reads

// y[m][l] = ref[m]*R1_l + B_l * dt*( f_{60m} + C_m ),  C_m = d60*C_{m-1} + S_{m-1}
// S = F @ D :  F[q][r] = clamp(lerp(ref[q],ref[q+1], r/60), 0)   (shared, LDS)
//              D[r][l] = d_l^(60-r)                               (in-register powers)
__global__ __launch_bounds__(WG_THREADS)
void srtm_wmma_kernel(const float* __restrict__ ref_tac,
                      const float* __restrict__ kparam,   // [3, L]: R1, k2, BPND
                      float* __restrict__ y,              // [61, L]
                      int L)
{
    __shared__ float Fsh[64 * FS];
    __shared__ float Ssh[64 * SS];

    const int tid    = threadIdx.x;
    const int wave   = tid >> 5;
    const int lane   = tid & 31;
    const int laneLo = lane & 15;
    const int laneHi = lane >> 4;          // 0 or 1
    const int wgBase = blockIdx.x * VOX_PER_WG;

    // ---- Stage F tile (64x64, rows q>=60 and cols r>=60 zero-padded) ----
    for (int idx = tid; idx < 64 * 64; idx += WG_THREADS) {
        const int q = idx >> 6, r = idx & 63;
        float v = 0.0f;
        if (q < 60 && r < 60) {
            const float a = ref_tac[q], b = ref_tac[q + 1];
            v = fmaxf(fmaf((float)r * DT, b - a, a), 0.0f);   // interp, clip >= 0
        }
        Fsh[q * FS + r] = v;
    }

    // ---- Per-lane voxel decay powers for the B operand (no memory traffic) ----
    int vox = wgBase + wave * 16 + laneLo;
    vox = vox < L ? vox : L - 1;
    const float kk2 = kparam[L + vox];
    const float bp  = kparam[2 * L + vox];
    const float d   = expf((-kk2 / (1.0f + bp)) * DT);        // exp(A*dt), in (0,1)
    const float d2  = d * d;
    const float d4  = d2 * d2;
    // kb iterates 56 -> 0; at kb: b = (d^(60-kb-aoff), d^(59-kb-aoff)), aoff = 2*laneHi
    float pwx = laneHi ? d2      : d4;                        // d^(4-aoff)
    float pwy = laneHi ? d       : d2 * d;                    // d^(3-aoff)
    const int aoff = laneHi ? 2 : 0;

    __syncthreads();

    // ---- S[64 x 16] = F[64 x 60] @ D[60 x 16] via V_WMMA_F32_16X16X4_F32 ----
    v8f acc0 = {}, acc1 = {}, acc2 = {}, acc3 = {};
    for (int kb = 56; kb >= 0; kb -= 4) {
        v2f b; b.x = pwx; b.y = pwy;
        const int col = kb + aoff;
        v2f a0 = { Fsh[(laneLo     ) * FS + col], Fsh[(laneLo     ) * FS + col + 1] };
        v2f a1 = { Fsh[(laneLo + 16) * FS + col], Fsh[(laneLo + 16) * FS + col + 1] };
        v2f a2 = { Fsh[(laneLo + 32) * FS + col], Fsh[(laneLo + 32) * FS + col + 1] };
        v2f a3 = { Fsh[(laneLo + 48) * FS + col], Fsh[(laneLo + 48) * FS + col + 1] };
        acc0 = __builtin_amdgcn_wmma_f32_16x16x4_f32(false, a0, false, b, (short)0, acc0, false, false);
        acc1 = __builtin_amdgcn_wmma_f32_16x16x4_f32(false, a1, false, b, (short)0, acc1, false, false);
        acc2 = __builtin_amdgcn_wmma_f32_16x16x4_f32(false, a2, false, b, (short)0, acc2, false, false);
        acc3 = __builtin_amdgcn_wmma_f32_16x16x4_f32(false, a3, false, b, (short)0, acc3, false, false);
        pwx *= d4; pwy *= d4;                                 // exponents grow by 4 as kb drops
    }

    // ---- Spill S to LDS in documented C/D layout: row = v + 8*laneHi (+16*mt) ----
    const int scol = wave * 16 + laneLo;
    #pragma unroll
    for (int v = 0; v < 8; ++v) {
        const int r0 = v + laneHi * 8;
        Ssh[(r0     ) * SS + scol] = acc0[v];
        Ssh[(r0 + 16) * SS + scol] = acc1[v];
        Ssh[(r0 + 32) * SS + scol] = acc2[v];
        Ssh[(r0 + 48) * SS + scol] = acc3[v];
    }
    __syncthreads();

    // ---- 60-step frame recurrence + epilogue (one thread per voxel) ----
    if (tid < VOX_PER_WG) {
        const int l = wgBase + tid;
        if (l < L) {
            const float R1  = kparam[l];
            const float kk2s = kparam[L + l];
            const float bps  = kparam[2 * L + l];
            const float As  = -kk2s / (1.0f + bps);
            const float Bs  = kk2s * (1.0f - R1 / (1.0f + bps));
            const float d60 = expf(As);                       // exp(A * dt * 60) = exp(A)
            float C = 0.0f;
            const float r0v = ref_tac[0];
            y[l] = fmaf(Bs, DT * fmaxf(r0v, 0.0f), r0v * R1); // conv_t[0] = dt*f_0
            for (int m = 1; m < T_FRAMES; ++m) {
                C = fmaf(d60, C, Ssh[(m - 1) * SS + tid]);
                const float rm   = ref_tac[m];
                const float conv = DT * (fmaxf(rm, 0.0f) + C);
                y[(size_t)m * L + l] = fmaf(Bs, conv, rm * R1);
            }
        }
    }
}

extern "C" void kernel_launch(void* const* d_in, const int* in_sizes, int n_in,
                              void* d_out, int out_size, void* d_ws, size_t ws_size,
                              hipStream_t stream) {
    // inputs: [0] t (int32,[61])  [1] ref_tac (f32,[61])  [2] k (f32,[3,L])  [3] n_tao (int)
    const float* ref_tac = (const float*)d_in[1];
    const float* kparam  = (const float*)d_in[2];
    float* y = (float*)d_out;
    const int L = in_sizes[2] / 3;
    const int grid = (L + VOX_PER_WG - 1) / VOX_PER_WG;   // 256 WGs for L=32768
    srtm_wmma_kernel<<<grid, WG_THREADS, 0, stream>>>(ref_tac, kparam, y, L);
}